// SelfAttentionLayer_4879082848975
// MI455X (gfx1250) — compile-verified
//
#include <hip/hip_runtime.h>
#include <float.h>

// ---------------------------------------------------------------------------
// Problem constants (from reference): B=32, DF=DO=2048, DK=DV=128, KER=4,
// STR=2, NC=(2048-4)/2+1=1023.  Query conv is kernel=stride=1 => rank-1 in k,
// so attention logits factor as a_q * s_c and the whole value path collapses
// to a 4-tap conv.  Output = [out (32*2048) | attn (32*2048*1023)] f32.
//
// Roofline: attn store stream = 268 MB (dominant, ~11.5us @ 23.3 TB/s).
// Emitted with nontemporal (TH=NT) stores since it exceeds the 192 MB L2 and
// is never re-read.  FFN = 0.54 GFLOP via v_wmma_f32_16x16x4_f32 (full f32).
// ---------------------------------------------------------------------------

#define Bq    32
#define DFq   2048
#define NCq   1023
#define NCPAD 1024

typedef __attribute__((ext_vector_type(2))) float v2f;
typedef __attribute__((ext_vector_type(8))) float v8f;

// ---------------- block reduction helpers (blockDim == 256, 8 waves) -------
__device__ __forceinline__ float blk_sum(float v, float* scratch) {
    int lane = threadIdx.x & 31, wv = threadIdx.x >> 5;
    #pragma unroll
    for (int o = 16; o; o >>= 1) v += __shfl_xor(v, o, 32);
    if (lane == 0) scratch[wv] = v;
    __syncthreads();
    float r = (threadIdx.x < 8) ? scratch[threadIdx.x] : 0.0f;
    if (wv == 0) {
        #pragma unroll
        for (int o = 4; o; o >>= 1) r += __shfl_xor(r, o, 32);
        if (lane == 0) scratch[0] = r;
    }
    __syncthreads();
    float out = scratch[0];
    __syncthreads();
    return out;
}

__device__ __forceinline__ float blk_max(float v, float* scratch) {
    int lane = threadIdx.x & 31, wv = threadIdx.x >> 5;
    #pragma unroll
    for (int o = 16; o; o >>= 1) v = fmaxf(v, __shfl_xor(v, o, 32));
    if (lane == 0) scratch[wv] = v;
    __syncthreads();
    float r = (threadIdx.x < 8) ? scratch[threadIdx.x] : -FLT_MAX;
    if (wv == 0) {
        #pragma unroll
        for (int o = 4; o; o >>= 1) r = fmaxf(r, __shfl_xor(r, o, 32));
        if (lane == 0) scratch[0] = r;
    }
    __syncthreads();
    float out = scratch[0];
    __syncthreads();
    return out;
}

__device__ __forceinline__ float blk_min(float v, float* scratch) {
    int lane = threadIdx.x & 31, wv = threadIdx.x >> 5;
    #pragma unroll
    for (int o = 16; o; o >>= 1) v = fminf(v, __shfl_xor(v, o, 32));
    if (lane == 0) scratch[wv] = v;
    __syncthreads();
    float r = (threadIdx.x < 8) ? scratch[threadIdx.x] : FLT_MAX;
    if (wv == 0) {
        #pragma unroll
        for (int o = 4; o; o >>= 1) r = fminf(r, __shfl_xor(r, o, 32));
        if (lane == 0) scratch[0] = r;
    }
    __syncthreads();
    float out = scratch[0];
    __syncthreads();
    return out;
}

// ---------------- K0: combine conv weights into 4-tap kernels --------------
// kqvw[0..3] = kq[t] = sum_k q_w[k]*k_w[k,t];  kqvw[4..7] = vw[t] = sum_v conv_w[v]*v_w[v,t]
__global__ void combine_weights_kernel(const float* __restrict__ q_w,
                                       const float* __restrict__ k_w,
                                       const float* __restrict__ v_w,
                                       const float* __restrict__ conv_w,
                                       float* __restrict__ kqvw) {
    int t = threadIdx.x;
    if (t < 4) {
        float s = 0.f;
        for (int k = 0; k < 128; ++k) s += q_w[k] * k_w[k * 4 + t];
        kqvw[t] = s;
    } else if (t < 8) {
        int tt = t - 4;
        float s = 0.f;
        for (int v = 0; v < 128; ++v) s += conv_w[v] * v_w[v * 4 + tt];
        kqvw[t] = s;
    }
}

// ---------------- K1: per-batch prep: LN(1e-5), gather, s/vv conv, min/max -
__global__ void prep_kernel(const float* __restrict__ feat,
                            const int*   __restrict__ idx,
                            const float* __restrict__ ln_w,
                            const float* __restrict__ ln_b,
                            const float* __restrict__ kqvw,
                            float* __restrict__ A_ln,
                            float* __restrict__ S,
                            float* __restrict__ VV,
                            float* __restrict__ smax,
                            float* __restrict__ smin) {
    __shared__ float sh[DFq];
    __shared__ float scratch[32];
    int b = blockIdx.x;
    const float* f = feat + (size_t)b * DFq;

    float p = 0.f;
    for (int q = threadIdx.x; q < DFq; q += blockDim.x) p += f[q];
    float mean = blk_sum(p, scratch) * (1.0f / DFq);
    float vp = 0.f;
    for (int q = threadIdx.x; q < DFq; q += blockDim.x) {
        float d = f[q] - mean; vp += d * d;
    }
    float var = blk_sum(vp, scratch) * (1.0f / DFq);
    float rstd = rsqrtf(var + 1e-5f);
    for (int q = threadIdx.x; q < DFq; q += blockDim.x)
        A_ln[(size_t)b * DFq + q] = (f[q] - mean) * rstd * ln_w[q] + ln_b[q];

    // gather shuffled row into LDS
    for (int q = threadIdx.x; q < DFq; q += blockDim.x) sh[q] = f[idx[q]];
    __syncthreads();

    float kq0 = kqvw[0], kq1 = kqvw[1], kq2 = kqvw[2], kq3 = kqvw[3];
    float vw0 = kqvw[4], vw1 = kqvw[5], vw2 = kqvw[6], vw3 = kqvw[7];
    float mx = -FLT_MAX, mn = FLT_MAX;
    for (int c = threadIdx.x; c < NCq; c += blockDim.x) {
        float x0 = sh[2 * c + 0], x1 = sh[2 * c + 1];
        float x2 = sh[2 * c + 2], x3 = sh[2 * c + 3];
        float s = x0 * kq0 + x1 * kq1 + x2 * kq2 + x3 * kq3;
        float v = x0 * vw0 + x1 * vw1 + x2 * vw2 + x3 * vw3;
        S [b * NCPAD + c] = s;
        VV[b * NCPAD + c] = v;
        mx = fmaxf(mx, s); mn = fminf(mn, s);
    }
    mx = blk_max(mx, scratch);
    mn = blk_min(mn, scratch);
    if (threadIdx.x == 0) { smax[b] = mx; smin[b] = mn; }
}

// ---------------- K2: softmax rows -> attn output + attention result -------
// grid (64, 32); block 256 (8 waves); each wave owns 4 q-rows of batch b.
// attn rows are written with nontemporal stores: 268 MB write-once stream,
// larger than L2, never re-read.
__global__ void attn_kernel(const float* __restrict__ A_ln,
                            const float* __restrict__ S,
                            const float* __restrict__ VV,
                            const float* __restrict__ smax,
                            const float* __restrict__ smin,
                            const float* __restrict__ feat,
                            float* __restrict__ attn,
                            float* __restrict__ out1) {
    __shared__ float s_sh[NCq];
    __shared__ float v_sh[NCq];
    int b = blockIdx.y;
    int lane = threadIdx.x & 31, wv = threadIdx.x >> 5;
    for (int c = threadIdx.x; c < NCq; c += blockDim.x) {
        s_sh[c] = S [b * NCPAD + c];
        v_sh[c] = VV[b * NCPAD + c];
    }
    __syncthreads();
    float mxs = smax[b], mns = smin[b];
    #pragma unroll
    for (int i = 0; i < 4; ++i) {
        int q = blockIdx.x * 32 + i * 8 + wv;
        float aq = A_ln[(size_t)b * DFq + q];
        float m  = (aq >= 0.f) ? aq * mxs : aq * mns;   // exact row max
        float denom = 0.f, num = 0.f;
        for (int c = lane; c < NCq; c += 32) {
            float e = __expf(aq * s_sh[c] - m);
            denom += e; num += e * v_sh[c];
        }
        #pragma unroll
        for (int o = 16; o; o >>= 1) {
            denom += __shfl_xor(denom, o, 32);
            num   += __shfl_xor(num,   o, 32);
        }
        float inv = 1.0f / denom;
        float* row = attn + ((size_t)b * DFq + q) * NCq;
        for (int c = lane; c < NCq; c += 32)
            __builtin_nontemporal_store(__expf(aq * s_sh[c] - m) * inv, row + c);
        if (lane == 0)
            out1[(size_t)b * DFq + q] = num * inv + feat[(size_t)b * DFq + q];
    }
}

// ---------------- K3: layernorm (eps 1e-6) ---------------------------------
__global__ void layernorm_kernel(const float* __restrict__ x,
                                 const float* __restrict__ w,
                                 const float* __restrict__ bb,
                                 float* __restrict__ y) {
    __shared__ float scratch[32];
    int b = blockIdx.x;
    const float* xr = x + (size_t)b * DFq;
    float p = 0.f;
    for (int q = threadIdx.x; q < DFq; q += blockDim.x) p += xr[q];
    float mean = blk_sum(p, scratch) * (1.0f / DFq);
    float vp = 0.f;
    for (int q = threadIdx.x; q < DFq; q += blockDim.x) {
        float d = xr[q] - mean; vp += d * d;
    }
    float var = blk_sum(vp, scratch) * (1.0f / DFq);
    float rstd = rsqrtf(var + 1e-6f);
    for (int q = threadIdx.x; q < DFq; q += blockDim.x)
        y[(size_t)b * DFq + q] = (xr[q] - mean) * rstd * w[q] + bb[q];
}

// ---------------- K4/K5: FFN GEMM via f32 WMMA 16x16x4 ---------------------
// Y[m,n] = sum_i X[m,i]*W[n,i] + bias[n]  (+ReLU | +res[m,n]).  M=32, N=K=2048.
// One wave: 32(M) x 16(N) tile, two v8f accumulators sharing B.  16 blocks x
// 8 waves cover N.  A layout (16x4 f32): lanes 0-15 K={0,1}, lanes 16-31
// K={2,3}.  B layout (4x16): VGPR0 rows K0/K2, VGPR1 rows K1/K3, N on lanes.
template <bool RELU, bool RES>
__global__ void gemm_wmma_kernel(const float* __restrict__ X,
                                 const float* __restrict__ W,
                                 const float* __restrict__ bias,
                                 const float* __restrict__ res,
                                 float* __restrict__ Y) {
    constexpr int K = DFq;
    int lane = threadIdx.x & 31;
    int wv   = threadIdx.x >> 5;
    int n0   = (blockIdx.x * 8 + wv) * 16;
    int half = lane >> 4;        // 0: K offsets {0,1}; 1: {2,3}
    int l    = lane & 15;

    v8f acc0 = {}; v8f acc1 = {};
    const float* xr0 = X + (size_t)l * K;
    const float* xr1 = X + (size_t)(l + 16) * K;
    const float* wr  = W + (size_t)(n0 + l) * K;

    #pragma unroll 4
    for (int k = 0; k < K; k += 4) {
        int ka = k + 2 * half;
        v2f a0, a1, bv;
        a0.x = xr0[ka]; a0.y = xr0[ka + 1];
        a1.x = xr1[ka]; a1.y = xr1[ka + 1];
        bv.x = wr[ka];  bv.y = wr[ka + 1];
        acc0 = __builtin_amdgcn_wmma_f32_16x16x4_f32(
            false, a0, false, bv, (short)0, acc0, false, false);
        acc1 = __builtin_amdgcn_wmma_f32_16x16x4_f32(
            false, a1, false, bv, (short)0, acc1, false, false);
    }

    int n = n0 + l;
    float bval = bias[n];
    #pragma unroll
    for (int r = 0; r < 8; ++r) {
        int m0 = r + 8 * half;      // rows for acc0
        float y0 = acc0[r] + bval;
        float y1 = acc1[r] + bval;  // rows m0+16 for acc1
        if (RELU) { y0 = fmaxf(y0, 0.f); y1 = fmaxf(y1, 0.f); }
        if (RES)  { y0 += res[(size_t)m0 * DFq + n];
                    y1 += res[(size_t)(m0 + 16) * DFq + n]; }
        Y[(size_t)m0 * DFq + n]        = y0;
        Y[(size_t)(m0 + 16) * DFq + n] = y1;
    }
}

// ---------------------------------------------------------------------------
extern "C" void kernel_launch(void* const* d_in, const int* in_sizes, int n_in,
                              void* d_out, int out_size, void* d_ws, size_t ws_size,
                              hipStream_t stream) {
    const float* feat   = (const float*)d_in[0];
    const int*   idx    = (const int*)  d_in[1];
    const float* ln_w   = (const float*)d_in[2];
    const float* ln_b   = (const float*)d_in[3];
    const float* q_w    = (const float*)d_in[4];
    const float* k_w    = (const float*)d_in[5];
    const float* v_w    = (const float*)d_in[6];
    const float* conv_w = (const float*)d_in[7];
    const float* bn_w   = (const float*)d_in[8];
    const float* bn_b   = (const float*)d_in[9];
    const float* w1_w   = (const float*)d_in[10];
    const float* w1_b   = (const float*)d_in[11];
    const float* w2_w   = (const float*)d_in[12];
    const float* w2_b   = (const float*)d_in[13];

    float* out_final = (float*)d_out;                 // [32*2048]
    float* attn_out  = out_final + Bq * DFq;          // [32*2048*1023]

    float* ws   = (float*)d_ws;
    float* kqvw = ws;                                 // 8
    float* smax = ws + 8;                             // 32
    float* smin = ws + 40;                            // 32
    float* A_ln = ws + 128;                           // 32*2048
    float* S    = A_ln + Bq * DFq;                    // 32*1024
    float* VV   = S    + Bq * NCPAD;                  // 32*1024
    float* OUT1 = VV   + Bq * NCPAD;                  // 32*2048
    float* H    = OUT1 + Bq * DFq;                    // 32*2048
    float* H1   = H    + Bq * DFq;                    // 32*2048

    combine_weights_kernel<<<1, 32, 0, stream>>>(q_w, k_w, v_w, conv_w, kqvw);

    prep_kernel<<<Bq, 256, 0, stream>>>(feat, idx, ln_w, ln_b, kqvw,
                                        A_ln, S, VV, smax, smin);

    attn_kernel<<<dim3(64, Bq), 256, 0, stream>>>(A_ln, S, VV, smax, smin,
                                                  feat, attn_out, OUT1);

    layernorm_kernel<<<Bq, 256, 0, stream>>>(OUT1, bn_w, bn_b, H);

    gemm_wmma_kernel<true,  false><<<16, 256, 0, stream>>>(H,  w1_w, w1_b, nullptr, H1);
    gemm_wmma_kernel<false, true ><<<16, 256, 0, stream>>>(H1, w2_w, w2_b, OUT1, out_final);
}